// GNNLayer_1958505087411
// MI455X (gfx1250) — compile-verified
//
#include <hip/hip_runtime.h>
#include <hip/hip_bf16.h>

// Problem constants (match the reference)
#define BB 8
#define NN 2048
#define DD 256
#define N16 (NN / 16)     // 128 row-tiles per batch

typedef __attribute__((ext_vector_type(2))) float v2f;
typedef __attribute__((ext_vector_type(8))) float v8f;

// fp32 WMMA: D(16x16) = A(16x4) * B(4x16) + C
// A layout (ISA 7.12.2, 32-bit A 16x4): lanes 0-15 -> K=k0,k0+1 ; lanes 16-31 -> K=k0+2,k0+3 ; M=lane%16
// B layout mirrored (N=lane%16, K striped the same way)
// C/D layout: VGPR r: lanes 0-15 -> M=r ; lanes 16-31 -> M=r+8 ; N=lane%16
static __device__ __forceinline__ v8f wmma_f32(v2f a, v2f b, v8f c) {
    return __builtin_amdgcn_wmma_f32_16x16x4_f32(
        /*neg_a=*/false, a, /*neg_b=*/false, b,
        /*c_mod=*/(short)0, c, /*reuse_a=*/false, /*reuse_b=*/false);
}

// ---------------------------------------------------------------------------
// Kernel 1: h[b,n,d] = sum_k x[b,n,k] * W[d,k] + bias[d]
// grid.x = B * N/16 blocks of 128 threads (4 waves); wave w covers d in [64w, 64w+64)
// ---------------------------------------------------------------------------
__global__ void __launch_bounds__(128)
h_gemm_kernel(const float* __restrict__ x, const float* __restrict__ W,
              const float* __restrict__ bias, float* __restrict__ h)
{
    const int lane  = threadIdx.x & 31;
    const int wave  = threadIdx.x >> 5;
    const int bi    = blockIdx.x;
    const int b     = bi / N16;
    const int rbase = (bi % N16) * 16;
    const int m     = lane & 15;
    const int khalf = (lane >> 4) * 2;       // 0 or 2
    const int dbase = wave * 64;

    const float* xrow = x + ((size_t)b * NN + rbase + m) * DD + khalf;

    v8f acc[4] = {};
    for (int k = 0; k < DD; k += 4) {
        v2f a = *(const v2f*)(xrow + k);
        #pragma unroll
        for (int t = 0; t < 4; ++t) {
            const int n = dbase + t * 16 + m;
            v2f bv = *(const v2f*)(W + (size_t)n * DD + k + khalf);
            acc[t] = wmma_f32(a, bv, acc[t]);
        }
    }

    const int rowadd = (lane >> 4) * 8;
    #pragma unroll
    for (int t = 0; t < 4; ++t) {
        const int col = dbase + t * 16 + m;
        const float bc = bias[col];
        #pragma unroll
        for (int r = 0; r < 8; ++r) {
            const int row = rbase + r + rowadd;
            h[((size_t)b * NN + row) * DD + col] = acc[t][r] + bc;
        }
    }
}

// ---------------------------------------------------------------------------
// Kernel 2: s1 = h @ a1, s2 = h @ a2  (one thread per (b,n) row)
// ---------------------------------------------------------------------------
__global__ void __launch_bounds__(256)
s12_kernel(const float* __restrict__ h, const float* __restrict__ a1,
           const float* __restrict__ a2, float* __restrict__ s1, float* __restrict__ s2)
{
    const int idx = blockIdx.x * blockDim.x + threadIdx.x;   // 0 .. B*N-1
    if (idx >= BB * NN) return;
    const float* hp = h + (size_t)idx * DD;
    float r1 = 0.f, r2 = 0.f;
    for (int d = 0; d < DD; ++d) {
        const float v = hp[d];
        r1 += v * a1[d];
        r2 += v * a2[d];
    }
    s1[idx] = r1;
    s2[idx] = r2;
}

// ---------------------------------------------------------------------------
// Kernel 3 (fused): h_new = normalize(att) @ h, single pass over adj.
//
// Block = 256 threads (8 waves) owns IROWS=64 rows of i (4 WMMA row-tiles) and
// all 256 d-columns (wave w -> cols [32w, 32w+32), 2 d-tiles). Per 64-wide j
// chunk the block stages the UNNORMALIZED att tile (sig * adj * mask_j) in LDS,
// accumulates per-row sums in registers, and each wave does 4x2 WMMAs per
// K-step reusing one pair of B loads across 4 i-tiles. At the end, row sums
// are reduced via LDS float atomics and the accumulators are scaled by
// mask_i / (mask_i * sum + 1e-8), which reproduces the reference exactly
// (masked rows -> 0).
// ---------------------------------------------------------------------------
#define IT 4                  // i-tiles per block
#define IROWS (IT * 16)       // 64 rows of i per block
#define TJ 64                 // j-chunk staged in LDS

__global__ void __launch_bounds__(256)
hnew_fused_kernel(const float* __restrict__ adj, const float* __restrict__ mask,
                  const float* __restrict__ h, const float* __restrict__ s1,
                  const float* __restrict__ s2, const float* __restrict__ att_bp,
                  float* __restrict__ out)
{
    __shared__ float att[IROWS * TJ];    // 16 KB: unnormalized att tile
    __shared__ float s1l[IROWS];         // s1[row] + att_b per block row
    __shared__ float dsum[IROWS];        // per-row denominators

    const int tid  = threadIdx.x;
    const int lane = tid & 31;
    const int wave = tid >> 5;
    const int bi   = blockIdx.x;
    const int b    = bi / (NN / IROWS);
    const int ib   = (bi % (NN / IROWS)) * IROWS;

    const float ab = att_bp[0];
    const float* s2b = s2   + (size_t)b * NN;
    const float* mb  = mask + (size_t)b * NN;
    const float* hb  = h    + (size_t)b * NN * DD;

    if (tid < IROWS) {
        dsum[tid] = 0.f;
        s1l[tid]  = s1[(size_t)b * NN + ib + tid] + ab;
    }

    const int m     = lane & 15;
    const int khalf = (lane >> 4) * 2;   // 0 or 2
    const int dbase = wave * 32;

    // att-fill assignment: thread -> (row = (tid>>6) + 4q, col = tid&63)
    const int fcol  = tid & 63;
    const int frow0 = tid >> 6;

    float psum[16];                      // per-thread partial row sums
    #pragma unroll
    for (int q = 0; q < 16; ++q) psum[q] = 0.f;

    v8f acc[IT][2] = {};

    for (int jc = 0; jc < NN; jc += TJ) {
        __syncthreads();                 // LDS tile reuse + covers init on jc==0
        #pragma unroll
        for (int q = 0; q < 16; ++q) {
            const int r = frow0 + 4 * q;
            const int j = jc + fcol;
            const float l  = s1l[r] + s2b[j];
            const float sg = 1.f / (1.f + __expf(-l));
            const float v  = sg * adj[((size_t)(b * NN + ib + r)) * NN + j] * mb[j];
            att[r * TJ + fcol] = v;
            psum[q] += v;
        }
        __syncthreads();

        #pragma unroll
        for (int ks = 0; ks < TJ / 4; ++ks) {
            const int k0 = ks * 4 + khalf;           // even -> 8B-aligned LDS reads
            const float* hr = hb + (size_t)(jc + k0) * DD;
            const int n0 = dbase + m;
            v2f bv0, bv1;
            bv0.x = hr[n0];        bv0.y = hr[DD + n0];
            bv1.x = hr[n0 + 16];   bv1.y = hr[DD + n0 + 16];
            #pragma unroll
            for (int it = 0; it < IT; ++it) {
                v2f a = *(const v2f*)&att[(it * 16 + m) * TJ + k0];
                acc[it][0] = wmma_f32(a, bv0, acc[it][0]);
                acc[it][1] = wmma_f32(a, bv1, acc[it][1]);
            }
        }
    }

    // reduce per-row denominators across the block (ds_add_f32 atomics)
    #pragma unroll
    for (int q = 0; q < 16; ++q)
        atomicAdd(&dsum[frow0 + 4 * q], psum[q]);
    __syncthreads();

    // scale accumulators by mask_i / (mask_i * sum + 1e-8) and store
    const int rowadd = (lane >> 4) * 8;
    #pragma unroll
    for (int it = 0; it < IT; ++it) {
        #pragma unroll
        for (int r = 0; r < 8; ++r) {
            const int rl  = it * 16 + r + rowadd;    // row within block
            const int row = ib + rl;
            const float mi = mb[row & (NN - 1)] * 0.f + mask[(size_t)b * NN + row];
            const float sc = mi / (mi * dsum[rl] + 1e-8f);
            const int col = dbase + m;
            out[((size_t)b * NN + row) * DD + col]      = acc[it][0][r] * sc;
            out[((size_t)b * NN + row) * DD + col + 16] = acc[it][1][r] * sc;
        }
    }
}

// ---------------------------------------------------------------------------
extern "C" void kernel_launch(void* const* d_in, const int* in_sizes, int n_in,
                              void* d_out, int out_size, void* d_ws, size_t ws_size,
                              hipStream_t stream)
{
    const float* x    = (const float*)d_in[0];
    const float* adj  = (const float*)d_in[1];
    const float* mask = (const float*)d_in[2];
    const float* W    = (const float*)d_in[3];
    const float* bias = (const float*)d_in[4];
    const float* a1   = (const float*)d_in[5];
    const float* a2   = (const float*)d_in[6];
    const float* attb = (const float*)d_in[7];
    float* outp = (float*)d_out;

    // workspace layout: h (B*N*D) | s1 (B*N) | s2 (B*N)   ~16.9 MB
    float* h  = (float*)d_ws;
    float* s1 = h  + (size_t)BB * NN * DD;
    float* s2 = s1 + (size_t)BB * NN;

    h_gemm_kernel<<<BB * N16, 128, 0, stream>>>(x, W, bias, h);
    s12_kernel<<<(BB * NN) / 256, 256, 0, stream>>>(h, a1, a2, s1, s2);
    hnew_fused_kernel<<<BB * (NN / IROWS), 256, 0, stream>>>(adj, mask, h, s1, s2, attb, outp);
}